// PrototypeLoss_40424232190320
// MI455X (gfx1250) — compile-verified
//
#include <hip/hip_runtime.h>

#define NUM_CLASSES 4096
#define BATCH       16384

typedef __attribute__((ext_vector_type(2))) float v2f;
typedef __attribute__((ext_vector_type(4))) float v4f;
typedef __attribute__((ext_vector_type(8))) float v8f;

// ---------------------------------------------------------------------------
// Wave32 min reduction via shuffle tree (cheap, only used once per row tile).
// ---------------------------------------------------------------------------
__device__ __forceinline__ float wave_min32(float v) {
#pragma unroll
    for (int off = 16; off >= 1; off >>= 1)
        v = fminf(v, __shfl_xor(v, off, 32));
    return v;
}

// ---------------------------------------------------------------------------
// Wave32 sum reduction via V_WMMA_F32_16X16X4_F32.
// A (16x4 f32, 2 VGPRs): VGPR0 = s  -> A[m][0] = s_m (lanes 0-15),
//                                      A[m][2] = s_{m+16} (lanes 16-31)
//                        VGPR1 = 0  -> A[m][1] = A[m][3] = 0
// B = all ones  =>  D[m][n] = s_m + s_{m+16}  for every n.
// D VGPR j holds M=j (lanes 0-15) / M=j+8 (lanes 16-31), so summing the 8
// accumulator VGPRs and adding across the half-waves yields the full wave sum
// in every lane, exactly in f32.
// EXEC must be all ones here (no divergence at call sites).
// ---------------------------------------------------------------------------
__device__ __forceinline__ float wave_sum_wmma(float s) {
    v2f a; a.x = s;    a.y = 0.0f;
    v2f b; b.x = 1.0f; b.y = 1.0f;
    v8f c = {};
    c = __builtin_amdgcn_wmma_f32_16x16x4_f32(
            /*neg_a=*/false, a, /*neg_b=*/false, b,
            /*c_mod=*/(short)0, c, /*reuse_a=*/false, /*reuse_b=*/false);
    float t = ((c[0] + c[1]) + (c[2] + c[3])) + ((c[4] + c[5]) + (c[6] + c[7]));
    t += __shfl_xor(t, 16, 32);
    return t;
}

// ---------------------------------------------------------------------------
// One block (256 threads = 8 waves) per row.  Each thread owns 16 columns as
// 4 x float4; chunk j covers columns [j*1024, (j+1)*1024), thread t takes the
// float4 at j*1024 + 4*t  -> every load instruction is a contiguous 512B
// burst per wave.  Data stays in registers between the min pass and the
// exp-sum pass, so HBM traffic is the theoretical minimum (read-once, NT).
// ---------------------------------------------------------------------------
__global__ __launch_bounds__(256)
void row_loss_kernel(const float* __restrict__ dist,
                     const int*   __restrict__ labels,
                     float*       __restrict__ row_out) {
    const int row  = blockIdx.x;
    const int t    = threadIdx.x;
    const int lane = t & 31;
    const int wave = t >> 5;

    const int    label = labels[row];
    const float* rp    = dist + (size_t)row * NUM_CLASSES;

    __shared__ float s_min[8];
    __shared__ float s_sum[8];
    __shared__ float s_correct;

    // ---- stream the row into registers (non-temporal, 128b, coalesced) ----
    v4f v[4];
#pragma unroll
    for (int j = 0; j < 4; ++j)
        v[j] = __builtin_nontemporal_load((const v4f*)(rp + j * 1024 + 4 * t));

    // ---- extract the label element; mask it with +inf so it is ignored ----
    const int lthr = (label & 1023) >> 2;   // owning thread within a chunk
    if (t == lthr) {
        const int lj = label >> 10;         // chunk
        const int lk = label & 3;           // element within float4
        s_correct = v[lj][lk];
        v[lj][lk] = __builtin_inff();       // excluded from min and exp-sum
    }

    // ---- pass 1: dmin = min over incorrect classes ----
    float m = v[0][0];
#pragma unroll
    for (int j = 0; j < 4; ++j)
#pragma unroll
        for (int k = 0; k < 4; ++k)
            m = fminf(m, v[j][k]);
    m = wave_min32(m);
    if (lane == 0) s_min[wave] = m;
    __syncthreads();
    float dmin = s_min[0];
#pragma unroll
    for (int w = 1; w < 8; ++w) dmin = fminf(dmin, s_min[w]);

    // ---- pass 2: sum exp(dmin - d) over incorrect classes ----
    // masked element: dmin - inf = -inf -> exp = 0, drops out naturally.
    float s = 0.0f;
#pragma unroll
    for (int j = 0; j < 4; ++j)
#pragma unroll
        for (int k = 0; k < 4; ++k)
            s += __expf(dmin - v[j][k]);

    s = wave_sum_wmma(s);                   // v_wmma_f32_16x16x4_f32
    if (lane == 0) s_sum[wave] = s;
    __syncthreads();
    float total = 0.0f;
#pragma unroll
    for (int w = 0; w < 8; ++w) total += s_sum[w];

    // min_dist_incorrect = -logsumexp(-d) = dmin - log(total)
    if (t == 0)
        row_out[row] = s_correct / (dmin - logf(total));
}

// ---------------------------------------------------------------------------
// Deterministic single-block mean over the 16384 per-row values.
// ---------------------------------------------------------------------------
__global__ __launch_bounds__(256)
void reduce_mean_kernel(const float* __restrict__ row_vals,
                        float*       __restrict__ out) {
    const int t    = threadIdx.x;
    const int lane = t & 31;
    const int wave = t >> 5;
    __shared__ float s_red[8];

    float acc = 0.0f;
    for (int i = t; i < BATCH; i += 256)
        acc += row_vals[i];
#pragma unroll
    for (int off = 16; off >= 1; off >>= 1)
        acc += __shfl_xor(acc, off, 32);
    if (lane == 0) s_red[wave] = acc;
    __syncthreads();
    if (t == 0) {
        float tot = 0.0f;
#pragma unroll
        for (int w = 0; w < 8; ++w) tot += s_red[w];
        out[0] = tot * (1.0f / (float)BATCH);
    }
}

extern "C" void kernel_launch(void* const* d_in, const int* in_sizes, int n_in,
                              void* d_out, int out_size, void* d_ws, size_t ws_size,
                              hipStream_t stream) {
    (void)in_sizes; (void)n_in; (void)out_size; (void)ws_size;
    const float* dist   = (const float*)d_in[0];
    const int*   labels = (const int*)d_in[1];
    float*       rowbuf = (float*)d_ws;     // 16384 floats = 64 KB scratch

    row_loss_kernel<<<BATCH, 256, 0, stream>>>(dist, labels, rowbuf);
    reduce_mean_kernel<<<1, 256, 0, stream>>>(rowbuf, (float*)d_out);
}